// StrucHIS_51582557225171
// MI455X (gfx1250) — compile-verified
//
#include <hip/hip_runtime.h>

#define NHN 100000
#define NTN 200000
#define NE  3200000
#define DIN 128
#define DOUT 64
#define NTY 5
#define NSEG (NHN * NTY)          // 500000
#define OUTN (NHN * NTY * DOUT)   // 32,000,000

typedef __attribute__((ext_vector_type(2))) float v2f;
typedef __attribute__((ext_vector_type(8))) float v8f;

// ---- order-preserving float<->uint encoding for atomicMax on signed floats ----
__device__ __forceinline__ unsigned f2o(float f) {
    unsigned u = __float_as_uint(f);
    return (u & 0x80000000u) ? ~u : (u | 0x80000000u);
}
__device__ __forceinline__ float o2f(unsigned u) {
    unsigned v = (u & 0x80000000u) ? (u & 0x7FFFFFFFu) : ~u;
    return __uint_as_float(v);
}

// ---- tiny setup: wa_l[128] = W^T a_l ; wea[64] = W_e a_e ; h_e[5] ----
__global__ void k_setup(const float* __restrict__ W, const float* __restrict__ a_l,
                        const float* __restrict__ W_e, const float* __restrict__ a_e,
                        const float* __restrict__ emb,
                        float* __restrict__ wa_l, float* __restrict__ wea,
                        float* __restrict__ h_e) {
    int t = threadIdx.x;
    if (t < DIN) {
        float s = 0.f;
        for (int j = 0; j < DOUT; ++j) s += W[j * DIN + t] * a_l[j];
        wa_l[t] = s;
    }
    if (t < 64) {
        float s = 0.f;
        for (int j = 0; j < 64; ++j) s += W_e[t * 64 + j] * a_e[j];
        wea[t] = s;
    }
    __syncthreads();
    if (t < NTY) {
        float s = 0.f;
        for (int k = 0; k < 64; ++k) s += emb[t * 64 + k] * wea[k];
        h_e[t] = s;
    }
}

// ---- h_l[i] = head_feature[i] . wa_l   (one wave32 per row, float4 loads) ----
__global__ __launch_bounds__(256) void k_hl(const float* __restrict__ head,
                                            const float* __restrict__ wa_l,
                                            float* __restrict__ h_l) {
    int w = (blockIdx.x * blockDim.x + threadIdx.x) >> 5;
    int lane = threadIdx.x & 31;
    if (w >= NHN) return;
    float4 v = *(const float4*)(head + (size_t)w * DIN + lane * 4);
    float4 c = *(const float4*)(wa_l + lane * 4);
    float p = v.x * c.x + v.y * c.y + v.z * c.z + v.w * c.w;
    for (int off = 16; off > 0; off >>= 1) p += __shfl_down(p, off, 32);
    if (lane == 0) h_l[w] = p;
}

// ---- h_tail GEMM via V_WMMA_F32_16X16X4_F32, fused h_r epilogue ----
// block = 128 threads = 4 waves; blockIdx = 16-row M tile; wave = 16-col N tile.
__global__ __launch_bounds__(128) void k_tail_proj(
        const float* __restrict__ tail, const float* __restrict__ W,
        const float* __restrict__ a_r,
        float* __restrict__ h_tail, float* __restrict__ h_r) {
    __shared__ float As[16][132];   // 16x128 A tile, padded
    __shared__ float Ws[64][132];   // full W (64x128), padded
    __shared__ float hr[16];

    const int tid = threadIdx.x;
    const int tile = blockIdx.x;

    const float* asrc = tail + (size_t)tile * 16 * DIN;
#pragma unroll
    for (int c = 0; c < 4; ++c) {
        int idx = c * 512 + tid * 4;
        int r = idx >> 7, col = idx & 127;
        float4 v = *(const float4*)(asrc + idx);
        *(float4*)&As[r][col] = v;
    }
#pragma unroll
    for (int c = 0; c < 16; ++c) {
        int idx = c * 512 + tid * 4;
        int r = idx >> 7, col = idx & 127;
        float4 v = *(const float4*)(W + idx);
        *(float4*)&Ws[r][col] = v;
    }
    if (tid < 16) hr[tid] = 0.f;
    __syncthreads();

    const int wv = tid >> 5;       // N tile index 0..3
    const int lane = tid & 31;
    const int half = lane >> 4;    // K half select per ISA A/B layout
    const int mn = lane & 15;      // M for A frag, N for B frag

    v8f acc = {};
#pragma unroll
    for (int k0 = 0; k0 < DIN; k0 += 4) {
        v2f a, b;
        a.x = As[mn][k0 + 2 * half];
        a.y = As[mn][k0 + 2 * half + 1];
        b.x = Ws[wv * 16 + mn][k0 + 2 * half];      // B[k][n] = W[n][k]
        b.y = Ws[wv * 16 + mn][k0 + 2 * half + 1];
        acc = __builtin_amdgcn_wmma_f32_16x16x4_f32(
            /*neg_a=*/false, a, /*neg_b=*/false, b,
            /*c_mod=*/(short)0, acc, /*reuse_a=*/false, /*reuse_b=*/false);
    }

    float arn = a_r[wv * 16 + mn];
#pragma unroll
    for (int r = 0; r < 8; ++r) {
        int m = r + 8 * half;                       // C/D layout: vgpr r, halves
        float val = acc[r];
        h_tail[((size_t)tile * 16 + m) * DOUT + wv * 16 + mn] = val;
        atomicAdd(&hr[m], val * arn);               // fused h_r = h_tail . a_r
    }
    __syncthreads();
    if (tid < 16) h_r[tile * 16 + tid] = hr[tid];
}

// ---- init: zero out, m = enc(-inf), denom = 0 ----
__global__ void k_init(float* __restrict__ out, unsigned* __restrict__ mmax,
                       float* __restrict__ denom) {
    int i = blockIdx.x * blockDim.x + threadIdx.x;
    if (i < OUTN) out[i] = 0.f;
    if (i < NSEG) { mmax[i] = 0x007FFFFFu; denom[i] = 0.f; }
}

// ---- edge pass 1: att = leaky_relu(h_l+h_r+h_e); segment max ----
__global__ __launch_bounds__(256) void k_att(
        const int* __restrict__ hi, const int* __restrict__ ti,
        const int* __restrict__ ty,
        const float* __restrict__ h_l, const float* __restrict__ h_r,
        const float* __restrict__ h_e,
        float* __restrict__ att, unsigned* __restrict__ mmax) {
    int e = blockIdx.x * blockDim.x + threadIdx.x;
    if (e >= NE) return;
    int h = hi[e], t = ty[e];
    float a = h_l[h] + h_r[ti[e]] + h_e[t];
    a = a > 0.f ? a : 0.2f * a;
    att[e] = a;
    atomicMax(&mmax[h * NTY + t], f2o(a));
}

// ---- edge pass 2: ex = exp(att - m[seg]); denom += ex (in-place on att) ----
__global__ __launch_bounds__(256) void k_exp(
        const int* __restrict__ hi, const int* __restrict__ ty,
        const unsigned* __restrict__ mmax,
        float* __restrict__ att, float* __restrict__ denom) {
    int e = blockIdx.x * blockDim.x + threadIdx.x;
    if (e >= NE) return;
    int seg = hi[e] * NTY + ty[e];
    float ex = __expf(att[e] - o2f(mmax[seg]));
    att[e] = ex;
    atomicAdd(&denom[seg], ex);
}

// ---- edge pass 3: w = ex/denom; out[seg,:] += w * h_tail[tail]  (wave/edge) ----
__global__ __launch_bounds__(256) void k_agg(
        const int* __restrict__ hi, const int* __restrict__ ti,
        const int* __restrict__ ty,
        const float* __restrict__ ex, const float* __restrict__ denom,
        const float* __restrict__ h_tail, float* __restrict__ out) {
    int w = (blockIdx.x * blockDim.x + threadIdx.x) >> 5;
    int lane = threadIdx.x & 31;
    if (w >= NE) return;
    int seg = hi[w] * NTY + ty[w];
    float wt = ex[w] / denom[seg];
    const float* src = h_tail + (size_t)ti[w] * DOUT;
    float* dst = out + (size_t)seg * DOUT;
    atomicAdd(&dst[lane], wt * src[lane]);
    atomicAdd(&dst[lane + 32], wt * src[lane + 32]);
}

extern "C" void kernel_launch(void* const* d_in, const int* in_sizes, int n_in,
                              void* d_out, int out_size, void* d_ws, size_t ws_size,
                              hipStream_t stream) {
    const float* head = (const float*)d_in[0];
    const float* tail = (const float*)d_in[1];
    const float* W    = (const float*)d_in[2];
    const float* W_e  = (const float*)d_in[3];
    const float* emb  = (const float*)d_in[4];
    const float* a_l  = (const float*)d_in[5];
    const float* a_r  = (const float*)d_in[6];
    const float* a_e  = (const float*)d_in[7];
    const int* hi = (const int*)d_in[8];
    const int* ti = (const int*)d_in[9];
    const int* ty = (const int*)d_in[10];
    float* out = (float*)d_out;

    // workspace layout (float offsets), ~69 MB total
    float* ws     = (float*)d_ws;
    float* wa_l   = ws;                       // 128
    float* wea    = ws + 128;                 // 64
    float* h_e    = ws + 192;                 // 8 (5 used)
    float* h_l    = ws + 200;                 // NHN
    float* h_r    = ws + 100200;              // NTN
    float* h_tail = ws + 300200;              // NTN*64 = 12,800,000
    float* attex  = ws + 13100200;            // NE
    unsigned* mmax = (unsigned*)(ws + 16300200); // NSEG
    float* denom  = ws + 16800200;            // NSEG

    k_setup<<<1, 128, 0, stream>>>(W, a_l, W_e, a_e, emb, wa_l, wea, h_e);
    k_init<<<OUTN / 256, 256, 0, stream>>>(out, mmax, denom);
    k_hl<<<(NHN * 32) / 256, 256, 0, stream>>>(head, wa_l, h_l);
    k_tail_proj<<<NTN / 16, 128, 0, stream>>>(tail, W, a_r, h_tail, h_r);
    k_att<<<NE / 256, 256, 0, stream>>>(hi, ti, ty, h_l, h_r, h_e, attex, mmax);
    k_exp<<<NE / 256, 256, 0, stream>>>(hi, ty, mmax, attex, denom);
    k_agg<<<(NE * 32) / 256, 256, 0, stream>>>(hi, ti, ty, attex, denom, h_tail, out);
}